// GCN_34376918237434
// MI455X (gfx1250) — compile-verified
//
#include <hip/hip_runtime.h>
#include <hip/hip_bf16.h>

#define N_NODES 100000
#define N_EDGES 1600000
#define N_GRAPHS 128
#define IN_DIM 128
#define HIDDEN 64
#define NUM_CLS 16

typedef __attribute__((ext_vector_type(2))) float v2f;
typedef __attribute__((ext_vector_type(8))) float v8f;

// ---------------------------------------------------------------- utilities
__global__ void zero_f32(float* p, int n) {
    int i = blockIdx.x * 256 + threadIdx.x;
    if (i < n) p[i] = 0.0f;
}

// int64 edges -> int32, degree accumulation (deg buffer must be zeroed first)
__global__ void prep_edges(const long long* __restrict__ ei,
                           int* __restrict__ src32, int* __restrict__ dst32,
                           float* __restrict__ deg) {
    int e = blockIdx.x * 256 + threadIdx.x;
    if (e >= N_EDGES) return;
    int s = (int)ei[e];
    int d = (int)ei[(size_t)N_EDGES + e];
    src32[e] = s;
    dst32[e] = d;
    atomicAdd(&deg[d], 1.0f);
}

__global__ void finalize_dinv(float* deg_dinv) {   // in: deg, out: dinv
    int i = blockIdx.x * 256 + threadIdx.x;
    if (i >= N_NODES) return;
    deg_dinv[i] = rsqrtf(deg_dinv[i] + 1.0f);
}

__global__ void compute_norm(const int* __restrict__ src, const int* __restrict__ dst,
                             const float* __restrict__ dinv, float* __restrict__ norm) {
    int e = blockIdx.x * 256 + threadIdx.x;
    if (e >= N_EDGES) return;
    norm[e] = dinv[src[e]] * dinv[dst[e]];
}

// ------------------------------------------------- WMMA GEMM + fused epilogue
// H   = (RELU_IN ? relu(A) : A) @ W            [N_NODES x HIDDEN]
// AGG = H * dinv^2 + bias   (self-loop + bias; edge kernel atomically adds)
//
// W is staged TRANSPOSED in LDS: sWT[col][k], row-padded to KDIM+2 so the
// B-fragment {W[k][col], W[k+1][col]} is one aligned ds_load_b64 (no packing
// moves before each v_wmma) and banks stay spread across the half-wave.
template<int KDIM, bool RELU_IN>
__global__ __launch_bounds__(256)
void gcn_gemm(const float* __restrict__ A, const float* __restrict__ W,
              const float* __restrict__ bias, const float* __restrict__ dinv,
              float* __restrict__ H, float* __restrict__ AGG) {
    constexpr int KP = KDIM + 2;                 // padded row length (words)
    __shared__ __align__(16) float sWT[HIDDEN * KP];

    for (int i = threadIdx.x; i < (KDIM * HIDDEN) / 4; i += 256) {
        float4 w4 = ((const float4*)W)[i];       // coalesced global read
        int base = i * 4;
        int k = base / HIDDEN;                   // HIDDEN divides 4-groups
        int n = base % HIDDEN;                   // n is a multiple of 4
        sWT[(n + 0) * KP + k] = w4.x;            // conflict-free scatter
        sWT[(n + 1) * KP + k] = w4.y;
        sWT[(n + 2) * KP + k] = w4.z;
        sWT[(n + 3) * KP + k] = w4.w;
    }
    __syncthreads();

    const int lane = threadIdx.x & 31;
    const int wave = threadIdx.x >> 5;
    const int m0   = (blockIdx.x * 8 + wave) * 16;
    if (m0 >= N_NODES) return;             // wave-uniform: EXEC stays all-ones

    const int mrow  = lane & 15;           // A row / B,C column within 16
    const int khalf = (lane >> 4) << 1;    // 0 or 2: which K pair this lane holds

    v8f acc[4] = {};                       // 16 x 64 tile (4 N-subtiles)
    const float*  Arow = A + (size_t)(m0 + mrow) * KDIM;
    const float*  Wcol = sWT + khalf;      // + col*KP + kb per use

    #pragma unroll 4
    for (int kb = 0; kb < KDIM; kb += 4) {
        float2 a2 = *(const float2*)(Arow + kb + khalf);
        if (RELU_IN) { a2.x = fmaxf(a2.x, 0.0f); a2.y = fmaxf(a2.y, 0.0f); }
        v2f a; a.x = a2.x; a.y = a2.y;
        #pragma unroll
        for (int nt = 0; nt < 4; ++nt) {
            int col = nt * 16 + mrow;
            float2 b2 = *(const float2*)(Wcol + col * KP + kb);  // ds_load_b64
            v2f b; b.x = b2.x; b.y = b2.y;
            acc[nt] = __builtin_amdgcn_wmma_f32_16x16x4_f32(
                false, a, false, b, (short)0, acc[nt], false, false);
        }
    }

    // epilogue: D layout -> VGPR r holds M = r + 8*(lane>=16), N = nt*16 + mrow
    const int rowAdd = (lane >> 4) * 8;
    float di2[8];
    #pragma unroll
    for (int r = 0; r < 8; ++r) {
        float d = dinv[m0 + rowAdd + r];
        di2[r] = d * d;
    }
    #pragma unroll
    for (int nt = 0; nt < 4; ++nt) {
        int col = nt * 16 + mrow;
        float bc = bias[col];
        #pragma unroll
        for (int r = 0; r < 8; ++r) {
            size_t idx = (size_t)(m0 + rowAdd + r) * HIDDEN + col;
            float v = acc[nt][r];
            H[idx]   = v;
            AGG[idx] = v * di2[r] + bc;
        }
    }
}

// ------------------------------------------------- edge aggregation (L2-bound)
// One wave per edge; lane handles 2 features (float2 gather + 2 f32 atomics).
__global__ __launch_bounds__(256)
void edge_agg(const int* __restrict__ src, const int* __restrict__ dst,
              const float* __restrict__ norm, const float* __restrict__ H,
              float* __restrict__ AGG) {
    int e = blockIdx.x * 8 + (threadIdx.x >> 5);
    if (e >= N_EDGES) return;
    int lane = threadIdx.x & 31;
    int s = src[e];
    int d = dst[e];
    float nrm = norm[e];
    float2 v = *(const float2*)(H + (size_t)s * HIDDEN + lane * 2);
    float* p = AGG + (size_t)d * HIDDEN + lane * 2;
    atomicAdd(p,     v.x * nrm);
    atomicAdd(p + 1, v.y * nrm);
}

// ------------------------------------------------- global mean pool (phase 1)
__global__ __launch_bounds__(256)
void pool_kernel(const float* __restrict__ AGG, const long long* __restrict__ batch,
                 float* __restrict__ sums, float* __restrict__ counts) {
    int i = blockIdx.x * 8 + (threadIdx.x >> 5);
    if (i >= N_NODES) return;
    int lane = threadIdx.x & 31;
    int g = (int)batch[i];
    float2 v = *(const float2*)(AGG + (size_t)i * HIDDEN + lane * 2);
    atomicAdd(&sums[g * HIDDEN + lane * 2],     v.x);
    atomicAdd(&sums[g * HIDDEN + lane * 2 + 1], v.y);
    if (lane == 0) atomicAdd(&counts[g], 1.0f);
}

// ------------------------------------------------- mean + classifier head
__global__ void final_linear(const float* __restrict__ sums, const float* __restrict__ counts,
                             const float* __restrict__ linW, const float* __restrict__ linb,
                             float* __restrict__ out) {
    int i = blockIdx.x * 256 + threadIdx.x;
    if (i >= N_GRAPHS * NUM_CLS) return;
    int g = i / NUM_CLS, c = i % NUM_CLS;
    float inv = 1.0f / fmaxf(counts[g], 1.0f);
    float acc = linb[c];
    #pragma unroll 8
    for (int h = 0; h < HIDDEN; ++h)
        acc += sums[g * HIDDEN + h] * inv * linW[h * NUM_CLS + c];
    out[i] = acc;
}

// ---------------------------------------------------------------- launcher
extern "C" void kernel_launch(void* const* d_in, const int* in_sizes, int n_in,
                              void* d_out, int out_size, void* d_ws, size_t ws_size,
                              hipStream_t stream) {
    const float*     x     = (const float*)d_in[0];
    const long long* eidx  = (const long long*)d_in[1];
    const long long* batch = (const long long*)d_in[2];
    const float* W1 = (const float*)d_in[3];
    const float* b1 = (const float*)d_in[4];
    const float* W2 = (const float*)d_in[5];
    const float* b2 = (const float*)d_in[6];
    const float* W3 = (const float*)d_in[7];
    const float* b3 = (const float*)d_in[8];
    const float* lW = (const float*)d_in[9];
    const float* lb = (const float*)d_in[10];
    float* out = (float*)d_out;

    // workspace layout (bytes, 1 KiB-aligned chunks)
    char* ws = (char*)d_ws;
    size_t off = 0;
    int*   src32 = (int*)(ws + off);   off += 6400000;            // 1.6M int32
    int*   dst32 = (int*)(ws + off);   off += 6400000;
    float* dinv  = (float*)(ws + off); off += 400384;             // deg -> dinv
    float* norm  = (float*)(ws + off); off += 6400000;
    float* H     = (float*)(ws + off); off += 25600000;           // node features
    float* AGG   = (float*)(ws + off); off += 25600000;           // aggregation
    float* sums  = (float*)(ws + off); off += 32768;              // 128x64
    float* cnts  = (float*)(ws + off); off += 1024;               // 128 (contig w/ sums)

    const int EB  = (N_EDGES + 255) / 256;     // thread-per-edge grids
    const int EW  = (N_EDGES + 7) / 8;         // wave-per-edge grids
    const int NB  = (N_NODES + 255) / 256;
    const int NW  = (N_NODES + 7) / 8;
    const int GB  = (N_NODES / 16 + 7) / 8;    // GEMM: 8 waves x 16 rows per block

    // 1. degree / dinv / norm prep
    zero_f32<<<NB, 256, 0, stream>>>(dinv, N_NODES);
    prep_edges<<<EB, 256, 0, stream>>>(eidx, src32, dst32, dinv);
    finalize_dinv<<<NB, 256, 0, stream>>>(dinv);
    compute_norm<<<EB, 256, 0, stream>>>(src32, dst32, dinv, norm);

    // 2. layer 1: h1 = x@W1 ; AGG = h1*dinv^2 + b1 ; edge scatter
    gcn_gemm<IN_DIM, false><<<GB, 256, 0, stream>>>(x,   W1, b1, dinv, H, AGG);
    edge_agg<<<EW, 256, 0, stream>>>(src32, dst32, norm, H, AGG);

    // 3. layer 2: relu fused into A-load
    gcn_gemm<HIDDEN, true><<<GB, 256, 0, stream>>>(AGG, W2, b2, dinv, H, AGG);
    edge_agg<<<EW, 256, 0, stream>>>(src32, dst32, norm, H, AGG);

    // 4. layer 3 (no relu after)
    gcn_gemm<HIDDEN, true><<<GB, 256, 0, stream>>>(AGG, W3, b3, dinv, H, AGG);
    edge_agg<<<EW, 256, 0, stream>>>(src32, dst32, norm, H, AGG);

    // 5. pool + head
    zero_f32<<<(N_GRAPHS * HIDDEN + N_GRAPHS + 255) / 256, 256, 0, stream>>>(
        sums, N_GRAPHS * HIDDEN + N_GRAPHS);
    pool_kernel<<<NW, 256, 0, stream>>>(AGG, batch, sums, cnts);
    final_linear<<<(N_GRAPHS * NUM_CLS + 255) / 256, 256, 0, stream>>>(
        sums, cnts, lW, lb, out);
}